// CAModule_10660108829344
// MI455X (gfx1250) — compile-verified
//
#include <hip/hip_runtime.h>
#include <hip/hip_bf16.h>

// CDNA5 / gfx1250, wave32.
// fp32 WMMA (V_WMMA_F32_16X16X4_F32) + async LDS staging (ASYNCcnt) with
// double-buffered LDS tiles.

typedef __attribute__((ext_vector_type(2))) float v2f;
typedef __attribute__((ext_vector_type(8))) float v8f;

#define CCH   512          // channels
#define NPIX  4096         // 64*64
#define LSTR  36           // LDS row stride (floats): 16B-aligned rows, conflict-free frags

// low 32 bits of a generic pointer to __shared__ = LDS (group-segment) byte offset
__device__ __forceinline__ unsigned lds_off(const void* p) {
    return (unsigned)(size_t)p;
}

// async copy 16B global -> LDS, per lane (tracked by ASYNCcnt)
__device__ __forceinline__ void async_b128(unsigned lds, const void* g) {
    asm volatile("global_load_async_to_lds_b128 %0, %1, off"
                 :: "v"(lds), "v"((unsigned long long)(size_t)g) : "memory");
}
// async copy 4B global -> LDS, per lane (allows transposing scatter)
__device__ __forceinline__ void async_b32(unsigned lds, const void* g) {
    asm volatile("global_load_async_to_lds_b32 %0, %1, off"
                 :: "v"(lds), "v"((unsigned long long)(size_t)g) : "memory");
}
#define WAIT_ASYNC(N) asm volatile("s_wait_asynccnt " #N ::: "memory")

// ---------------------------------------------------------------------------
// Kernel 1: energy[b] = q[b] * q[b]^T
// Block: 256 thr (8 waves), 64x128 output tile, K chunks of 32, double-buffered.
// Wave w: subtile row sr=w&3, subtile cols scB..scB+3 (scB = (w>>2)*4) ->
// one shared A fragment feeds 4 WMMAs per k-step.
// ---------------------------------------------------------------------------
__global__ __launch_bounds__(256)
void ca_energy_kernel(const float* __restrict__ x, float* __restrict__ energy) {
    const int b  = blockIdx.y;
    const int ti = blockIdx.x >> 2;      // 0..7 : 64-row i tile
    const int tj = blockIdx.x & 3;       // 0..3 : 128-col j tile
    const int i0 = ti * 64, j0 = tj * 128;

    __shared__ float lds_a[2][64 * LSTR];    // q rows i0..+63  x 32 k
    __shared__ float lds_b[2][128 * LSTR];   // q rows j0..+127 x 32 k

    const int tid  = threadIdx.x;
    const int wave = tid >> 5;
    const int lane = tid & 31;
    const int lr   = lane & 15;
    const int lh   = lane >> 4;

    const int sr  = wave & 3;            // subtile row 0..3
    const int scB = (wave >> 2) * 4;     // subtile col base: 0 or 4

    v8f acc[4] = {};

    const float* qb = x + (size_t)b * CCH * NPIX;

    const int lrow = tid >> 3;           // 0..31
    const int lcol = (tid & 7) * 4;      // 0..28

    const float* gA0 = qb + (size_t)(i0      + lrow) * NPIX + lcol;
    const float* gA1 = qb + (size_t)(i0 + 32 + lrow) * NPIX + lcol;
    const float* gB0 = qb + (size_t)(j0      + lrow) * NPIX + lcol;
    const float* gB1 = gB0 + (size_t)32 * NPIX;
    const float* gB2 = gB0 + (size_t)64 * NPIX;
    const float* gB3 = gB0 + (size_t)96 * NPIX;

    auto stage = [&](int buf, int k0) {  // 6 async b128 per thread = full 64x32 + 128x32
        async_b128(lds_off(&lds_a[buf][lrow        * LSTR + lcol]), gA0 + k0);
        async_b128(lds_off(&lds_a[buf][(32 + lrow) * LSTR + lcol]), gA1 + k0);
        async_b128(lds_off(&lds_b[buf][lrow        * LSTR + lcol]), gB0 + k0);
        async_b128(lds_off(&lds_b[buf][(32 + lrow) * LSTR + lcol]), gB1 + k0);
        async_b128(lds_off(&lds_b[buf][(64 + lrow) * LSTR + lcol]), gB2 + k0);
        async_b128(lds_off(&lds_b[buf][(96 + lrow) * LSTR + lcol]), gB3 + k0);
    };

    stage(0, 0);
    for (int t = 0; t < NPIX / 32; ++t) {
        const int cur = t & 1;
        if (t + 1 < NPIX / 32) {
            stage(1 - cur, (t + 1) * 32);   // prefetch next chunk into other buffer
            WAIT_ASYNC(6);                  // current chunk (first 6 in flight) done
        } else {
            WAIT_ASYNC(0);
        }
        __syncthreads();

        const float* pa = &lds_a[cur][(sr * 16 + lr) * LSTR + 2 * lh];
        const float* pb = &lds_b[cur][lr * LSTR + 2 * lh];   // B = q^T: B[k][j] = q[j][k]
#pragma unroll
        for (int kk = 0; kk < 32; kk += 4) {
            v2f a = *(const v2f*)(pa + kk);
#pragma unroll
            for (int s = 0; s < 4; ++s) {
                v2f bb = *(const v2f*)(pb + (scB + s) * 16 * LSTR + kk);
                acc[s] = __builtin_amdgcn_wmma_f32_16x16x4_f32(
                    false, a, false, bb, (short)0, acc[s], false, false);
            }
        }
        __syncthreads();
    }

    // C/D layout: VGPR r -> (M = 16*sr + r + 8*lh, N = 16*sc + lr)
    float* eb = energy + (size_t)b * CCH * CCH;
    const int mi = i0 + sr * 16 + 8 * lh;
#pragma unroll
    for (int s = 0; s < 4; ++s) {
        const int nj = j0 + (scB + s) * 16 + lr;
#pragma unroll
        for (int r = 0; r < 8; ++r)
            eb[(size_t)(mi + r) * CCH + nj] = acc[s][r];
    }
}

// ---------------------------------------------------------------------------
// Kernel 2: in-place row softmax of (rowmax - e)  ==  exp(rowmin - e)/sum
// One wave per row of 512; lane-register row + wave32 shfl reductions.
// ---------------------------------------------------------------------------
__global__ __launch_bounds__(256)
void ca_softmax_kernel(float* __restrict__ att) {
    const int row  = blockIdx.x * 8 + (threadIdx.x >> 5);
    const int lane = threadIdx.x & 31;
    float* p = att + (size_t)row * CCH;

    float v[16];
    float rmin = 3.4e38f;
#pragma unroll
    for (int t = 0; t < 16; ++t) {
        v[t] = p[t * 32 + lane];
        rmin = fminf(rmin, v[t]);
    }
#pragma unroll
    for (int m = 16; m >= 1; m >>= 1) rmin = fminf(rmin, __shfl_xor(rmin, m, 32));

    float s = 0.0f;
#pragma unroll
    for (int t = 0; t < 16; ++t) {
        v[t] = __expf(rmin - v[t]);
        s += v[t];
    }
#pragma unroll
    for (int m = 16; m >= 1; m >>= 1) s += __shfl_xor(s, m, 32);

    const float inv = 1.0f / s;
#pragma unroll
    for (int t = 0; t < 16; ++t) p[t * 32 + lane] = v[t] * inv;
}

// ---------------------------------------------------------------------------
// Kernel 3: out[b] = beta * (att[b] * q[b])
// Block: 256 thr, 64x128 out tile, K=512 in chunks of 32, double-buffered.
// q chunk is transposed *inside* the async copy: per-lane b32 scatter into LDS,
// so B fragments are contiguous 2-float reads (lds_bt[n][k]).
// ---------------------------------------------------------------------------
__global__ __launch_bounds__(256)
void ca_out_kernel(const float* __restrict__ x, const float* __restrict__ att,
                   const float* __restrict__ beta, float* __restrict__ out) {
    const int b  = blockIdx.y;
    const int ti = blockIdx.x >> 5;      // 0..7  : 64-row M tile
    const int tn = blockIdx.x & 31;      // 0..31 : 128-col N tile
    const int i0 = ti * 64, n0 = tn * 128;

    __shared__ float lds_a [2][64 * LSTR];   // att rows i0..+63 x 32 k
    __shared__ float lds_bt[2][128 * LSTR];  // q^T: 128 n-rows x 32 k-cols

    const int tid  = threadIdx.x;
    const int wave = tid >> 5;
    const int lane = tid & 31;
    const int lr   = lane & 15;
    const int lh   = lane >> 4;

    const int sr  = wave & 3;
    const int scB = (wave >> 2) * 4;

    v8f acc[4] = {};

    const float* qb = x   + (size_t)b * CCH * NPIX;
    const float* ab = att + (size_t)b * CCH * CCH;

    const int lrow = tid >> 3;           // 0..31
    const int lcol = (tid & 7) * 4;      // 0..28
    const int kq   = tid >> 3;           // 0..31 : k row for q staging
    const int nq   = tid & 7;            // 0..7  : n column base for q staging

    const float* gA0 = ab + (size_t)(i0      + lrow) * CCH + lcol;
    const float* gA1 = ab + (size_t)(i0 + 32 + lrow) * CCH + lcol;

    auto stage = [&](int buf, int k0) {  // 2 b128 (att) + 16 transposing b32 (q) = 18
        async_b128(lds_off(&lds_a[buf][lrow        * LSTR + lcol]), gA0 + k0);
        async_b128(lds_off(&lds_a[buf][(32 + lrow) * LSTR + lcol]), gA1 + k0);
        const float* gq = qb + (size_t)(k0 + kq) * NPIX + n0 + nq;
#pragma unroll
        for (int i2 = 0; i2 < 16; ++i2)
            async_b32(lds_off(&lds_bt[buf][(nq + 8 * i2) * LSTR + kq]), gq + 8 * i2);
    };

    stage(0, 0);
    for (int t = 0; t < CCH / 32; ++t) {
        const int cur = t & 1;
        if (t + 1 < CCH / 32) {
            stage(1 - cur, (t + 1) * 32);
            WAIT_ASYNC(18);
        } else {
            WAIT_ASYNC(0);
        }
        __syncthreads();

        const float* pa = &lds_a [cur][(sr * 16 + lr) * LSTR + 2 * lh];
        const float* pb = &lds_bt[cur][lr * LSTR + 2 * lh];
#pragma unroll
        for (int kk = 0; kk < 32; kk += 4) {
            v2f a = *(const v2f*)(pa + kk);
#pragma unroll
            for (int s = 0; s < 4; ++s) {
                v2f bb = *(const v2f*)(pb + (scB + s) * 16 * LSTR + kk);
                acc[s] = __builtin_amdgcn_wmma_f32_16x16x4_f32(
                    false, a, false, bb, (short)0, acc[s], false, false);
            }
        }
        __syncthreads();
    }

    const float bscale = beta[0];
    float* ob = out + (size_t)b * CCH * NPIX;
    const int mi = i0 + sr * 16 + 8 * lh;
#pragma unroll
    for (int s = 0; s < 4; ++s) {
        const int nn = n0 + (scB + s) * 16 + lr;
#pragma unroll
        for (int r = 0; r < 8; ++r)
            ob[(size_t)(mi + r) * NPIX + nn] = bscale * acc[s][r];
    }
}

// ---------------------------------------------------------------------------
extern "C" void kernel_launch(void* const* d_in, const int* in_sizes, int n_in,
                              void* d_out, int out_size, void* d_ws, size_t ws_size,
                              hipStream_t stream) {
    (void)in_sizes; (void)n_in; (void)out_size; (void)ws_size;
    const float* x    = (const float*)d_in[0];   // [16,512,64,64] fp32
    const float* beta = (const float*)d_in[1];   // [1] fp32
    float*       out  = (float*)d_out;           // [16,512,64,64] fp32
    float*       att  = (float*)d_ws;            // 16*512*512 fp32 = 16.8 MB scratch

    // energy = q q^T : 8 x 4 tiles of 64x128 per batch
    ca_energy_kernel<<<dim3(32, 16), 256, 0, stream>>>(x, att);
    // row softmax of (rowmax - e), in place: 8192 rows, 8 per block
    ca_softmax_kernel<<<dim3(1024), 256, 0, stream>>>(att);
    // out = beta * att q : 8 x 32 tiles of 64x128 per batch
    ca_out_kernel<<<dim3(256, 16), 256, 0, stream>>>(x, att, beta, out);
}